// SupConLossKCLTSC_3453153706408
// MI455X (gfx1250) — compile-verified
//
#include <hip/hip_runtime.h>
#include <math.h>

// MI455X (gfx1250) fused SupCon-KCL loss.
// Gram matrix 4096x4096x128 (4.3 GFLOP) over a 2MB operand resident in the
// 192MB L2; the 67MB logits matrix is never materialized (online LSE).
// f32 WMMA (V_WMMA_F32_16X16X4_F32) keeps exact f32 while driving the CDNA5
// matrix pipe. B fragments are batch-loaded (32 outstanding global_load_b64)
// so the wmma chain isn't latency-bound, and the softmax epilogue is pure
// per-lane VALU (no in-loop cross-lane ds traffic).

typedef __attribute__((ext_vector_type(2))) float v2f;
typedef __attribute__((ext_vector_type(8))) float v8f;

#define TWO_B 4096
#define Bn    2048
#define Dn    128
#define INVT  (1.0f/0.07f)
#define NEGC  (-1e30f)

// ---------------- threefry2x32 (JAX partitionable path, key=(0,42)) --------
__device__ __forceinline__ unsigned rotl32(unsigned x, int n){ return (x<<n)|(x>>(32-n)); }

__device__ __forceinline__ unsigned threefry_out0(unsigned ctr){
  const unsigned ks0 = 0u, ks1 = 42u, ks2 = 0x1BD11BDAu ^ 42u;
  unsigned x0 = ks0;        // counter hi = 0 (size 2^24 < 2^32)
  unsigned x1 = ctr + ks1;  // counter lo = flat index
  #define TFR(r) { x0 += x1; x1 = rotl32(x1,(r)); x1 ^= x0; }
  TFR(13) TFR(15) TFR(26) TFR(6)  x0 += ks1; x1 += ks2 + 1u;
  TFR(17) TFR(29) TFR(16) TFR(24) x0 += ks2; x1 += ks0 + 2u;
  TFR(13) TFR(15) TFR(26) TFR(6)  x0 += ks0; x1 += ks1 + 3u;
  TFR(17) TFR(29) TFR(16) TFR(24) x0 += ks1; x1 += ks2 + 4u;
  TFR(13) TFR(15) TFR(26) TFR(6)  x0 += ks2; x1 += ks0 + 5u;
  #undef TFR
  return x0;
}

__device__ __forceinline__ float gumbel_at(unsigned m){
  unsigned bits = threefry_out0(m);
  float f = __uint_as_float((bits >> 9) | 0x3F800000u) - 1.0f; // [0,1)
  float u = fmaxf(f + 1e-9f, 1e-9f);                           // jax uniform(1e-9, 1.0)
  return -logf(-logf(u));
}

// ---------------- kernel A: pack contrast features ------------------------
__global__ void build_xb(const float* __restrict__ feat, float* __restrict__ xb){
  int idx = blockIdx.x * blockDim.x + threadIdx.x;
  if (idx >= TWO_B * Dn) return;
  int i = idx >> 7, d = idx & 127;
  int b = i & (Bn - 1);
  int v = i >> 11;                       // view 0 for rows <2048, else view 1
  xb[idx] = feat[(b * 2 + v) * Dn + d];
}

// ---------------- kernel B: Gumbel top-3 + positive numerator -------------
__global__ void __launch_bounds__(256)
topk_num(const float* __restrict__ xb, const int* __restrict__ labels,
         float* __restrict__ num, float* __restrict__ npos){
  const int i = blockIdx.x;
  const int tid = threadIdx.x;
  const int partner = i ^ 2048;
  const int myl = labels[i & (Bn - 1)];

  float v0 = NEGC, v1 = NEGC, v2 = NEGC;
  int   i0 = 0x7fffffff, i1 = 0x7fffffff, i2 = 0x7fffffff;
  for (int j = tid; j < TWO_B; j += 256){
    if (j == i || j == partner) continue;
    if (labels[j & (Bn - 1)] != myl) continue;
    float g = gumbel_at((unsigned)(i * TWO_B + j));
    if (g > v0 || (g == v0 && j < i0)) { v2=v1;i2=i1; v1=v0;i1=i0; v0=g;i0=j; }
    else if (g > v1 || (g == v1 && j < i1)) { v2=v1;i2=i1; v1=g;i1=j; }
    else if (g > v2 || (g == v2 && j < i2)) { v2=g;i2=j; }
  }

  __shared__ float sv[256 * 3];
  __shared__ int   si[256 * 3];
  sv[tid*3+0]=v0; sv[tid*3+1]=v1; sv[tid*3+2]=v2;
  si[tid*3+0]=i0; si[tid*3+1]=i1; si[tid*3+2]=i2;
  for (int stride = 128; stride > 0; stride >>= 1){
    __syncthreads();
    if (tid < stride){
      float a0=sv[tid*3],a1=sv[tid*3+1],a2=sv[tid*3+2];
      int   A0=si[tid*3],A1=si[tid*3+1],A2=si[tid*3+2];
      #pragma unroll
      for (int t = 0; t < 3; ++t){
        float g = sv[(tid+stride)*3+t]; int j = si[(tid+stride)*3+t];
        if (g > a0 || (g == a0 && j < A0)) { a2=a1;A2=A1; a1=a0;A1=A0; a0=g;A0=j; }
        else if (g > a1 || (g == a1 && j < A1)) { a2=a1;A2=A1; a1=g;A1=j; }
        else if (g > a2 || (g == a2 && j < A2)) { a2=g;A2=j; }
      }
      sv[tid*3]=a0; sv[tid*3+1]=a1; sv[tid*3+2]=a2;
      si[tid*3]=A0; si[tid*3+1]=A1; si[tid*3+2]=A2;
    }
  }
  __syncthreads();

  // dot(x_i, x_j) for partner + up to 3 sampled positives (4 waves)
  __shared__ float dres[4];
  int g = tid >> 5, lane = tid & 31;
  if (g < 4){
    int jsel; bool ok;
    if (g == 0){ jsel = partner; ok = true; }
    else       { jsel = si[g-1]; ok = (sv[g-1] > NEGC * 0.5f); }
    float s = 0.f;
    if (ok){
      const float* xi = xb + i * Dn;
      const float* xj = xb + jsel * Dn;
      for (int d = lane; d < Dn; d += 32) s += xi[d] * xj[d];
    }
    #pragma unroll
    for (int off = 16; off; off >>= 1) s += __shfl_xor(s, off);
    if (lane == 0) dres[g] = ok ? s : 0.f;
  }
  __syncthreads();
  if (tid == 0){
    int nv = 0;
    #pragma unroll
    for (int t = 0; t < 3; ++t) if (sv[t] > NEGC * 0.5f) nv++;
    num[i]  = (dres[0] + dres[1] + dres[2] + dres[3]) * INVT;
    npos[i] = (float)(nv + 1);
  }
}

// ---------------- kernel C: f32-WMMA Gram strip + online LSE + loss -------
__global__ void __launch_bounds__(128)
lse_loss(const float* __restrict__ xb, const float* __restrict__ num,
         const float* __restrict__ npos, float* __restrict__ rowloss){
  const int i0   = blockIdx.x * 16;
  const int tid  = threadIdx.x;     // 128 = 4 waves
  const int wid  = tid >> 5;
  const int lane = tid & 31;
  const int hl   = lane & 15;
  const int hi16 = lane >> 4;       // 0 or 1
  const int koff = hi16 * 2;        // ISA 7.12.2: lanes16-31 hold K=2,3

  __shared__ float As[16 * Dn];
  for (int t = tid; t < 16 * Dn; t += 128){
    int r = t >> 7, d = t & 127;
    As[t] = xb[(i0 + r) * Dn + d];
  }
  __syncthreads();

  // Preload A fragments for all 32 K-steps (v2f each) into registers.
  v2f afr[32];
  #pragma unroll
  for (int k = 0; k < 32; ++k)
    afr[k] = *(const v2f*)(&As[hl * Dn + 4 * k + koff]);

  // Per-lane online softmax stats for this lane's column slice of each row.
  float mx[8], sm[8];
  #pragma unroll
  for (int r = 0; r < 8; ++r){ mx[r] = NEGC; sm[r] = 0.f; }

  const int irowlo = i0 + hi16 * 8;   // C/D layout: vgpr r -> rows r / r+8

  for (int jt = wid; jt < 256; jt += 4){
    const int j0 = jt * 16;
    const float* bp = xb + (j0 + hl) * Dn + koff;   // B[k][n] = xb[j0+n][k]

    // Prefetch this wave's next column tile toward the WGP cache.
    if (jt + 4 < 256)
      __builtin_prefetch(xb + ((jt + 4) * 16 + hl) * Dn + koff, 0, 0);

    // Batch-load all 32 B fragments: 32 outstanding global_load_b64.
    v2f bfr[32];
    #pragma unroll
    for (int k = 0; k < 32; ++k)
      bfr[k] = *(const v2f*)(bp + 4 * k);

    v8f acc = {0.f,0.f,0.f,0.f,0.f,0.f,0.f,0.f};
    #pragma unroll
    for (int k = 0; k < 32; ++k)
      acc = __builtin_amdgcn_wmma_f32_16x16x4_f32(
          false, afr[k], false, bfr[k], (short)0, acc, false, false);

    const int jcol = j0 + hl;
    #pragma unroll
    for (int r = 0; r < 8; ++r){
      const int irow = irowlo + r;
      if (jcol != irow){                       // diagonal excluded via EXEC
        float a = acc[r] * INVT;
        float nm = fmaxf(mx[r], a);
        sm[r] = sm[r] * __expf(mx[r] - nm) + __expf(a - nm);
        mx[r] = nm;
      }
    }
  }

  // Merge the 16 per-lane stats of each half-wave (rows r / r+8).
  #pragma unroll
  for (int r = 0; r < 8; ++r){
    float m = mx[r], s = sm[r];
    #pragma unroll
    for (int d = 1; d < 16; d <<= 1){
      float mo = __shfl_xor(m, d);
      float so = __shfl_xor(s, d);
      float nm = fmaxf(m, mo);
      s = s * __expf(m - nm) + so * __expf(mo - nm);
      m = nm;
    }
    mx[r] = m; sm[r] = s;
  }

  __shared__ float wmx[4][16];
  __shared__ float wsm[4][16];
  if (hl == 0){
    int rbase = hi16 * 8;
    #pragma unroll
    for (int r = 0; r < 8; ++r){ wmx[wid][rbase + r] = mx[r]; wsm[wid][rbase + r] = sm[r]; }
  }
  __syncthreads();
  if (tid < 16){
    float m = wmx[0][tid];
    #pragma unroll
    for (int w = 1; w < 4; ++w) m = fmaxf(m, wmx[w][tid]);
    float s = 0.f;
    #pragma unroll
    for (int w = 0; w < 4; ++w) s += wsm[w][tid] * __expf(wmx[w][tid] - m);
    float lse = m + logf(s);
    int i = i0 + tid;
    rowloss[i] = lse - num[i] / npos[i];   // = -(T/BT)*mean_log_prob_pos
  }
}

// ---------------- kernel D: deterministic mean ----------------------------
__global__ void reduce_mean(const float* __restrict__ rowloss, float* __restrict__ out){
  __shared__ float red[256];
  int tid = threadIdx.x;
  float s = 0.f;
  for (int j = tid; j < TWO_B; j += 256) s += rowloss[j];
  red[tid] = s; __syncthreads();
  for (int st = 128; st; st >>= 1){ if (tid < st) red[tid] += red[tid + st]; __syncthreads(); }
  if (tid == 0) out[0] = red[0] * (1.0f / TWO_B);
}

extern "C" void kernel_launch(void* const* d_in, const int* in_sizes, int n_in,
                              void* d_out, int out_size, void* d_ws, size_t ws_size,
                              hipStream_t stream){
  const float* feat   = (const float*)d_in[0];
  const int*   labels = (const int*)d_in[1];
  // d_in[2] (prototypes) cancels exactly in the loss -> unused.
  float* out = (float*)d_out;

  float* xb      = (float*)d_ws;            // 4096*128 f32 (2 MB)
  float* num     = xb + TWO_B * Dn;         // 4096
  float* npos    = num + TWO_B;             // 4096
  float* rowloss = npos + TWO_B;            // 4096

  build_xb   <<<(TWO_B * Dn + 255) / 256, 256, 0, stream>>>(feat, xb);
  topk_num   <<<TWO_B, 256, 0, stream>>>(xb, labels, num, npos);
  lse_loss   <<<TWO_B / 16, 128, 0, stream>>>(xb, num, npos, rowloss);
  reduce_mean<<<1, 256, 0, stream>>>(rowloss, out);
}